// Low_Rank_TBRNN_76020921140337
// MI455X (gfx1250) — compile-verified
//
#include <hip/hip_runtime.h>
#include <hip/hip_bf16.h>
#include <math.h>

// Problem constants (fixed by the reference)
#define BSZ     256
#define TLEN    1024
#define INDIM   128
#define OUTDIM  64
#define HDIM    512
#define RDIM    8
#define NOISE_STD 0.05f
#define TAU       0.2f
#define INV_H2    (1.0f / (512.0f * 512.0f))

typedef __attribute__((ext_vector_type(2))) float v2f;
typedef __attribute__((ext_vector_type(8))) float v8f;

static __device__ __forceinline__ v8f wmma_f32_16x16x4(v2f a, v2f b, v8f c) {
    // D = A(16x4,f32) x B(4x16,f32) + C(16x16,f32)
    return __builtin_amdgcn_wmma_f32_16x16x4_f32(false, a, false, b, (short)0, c,
                                                 false, false);
}

// ---------------------------------------------------------------------------
// Kernel A: input projection  proj[b,t,h] = u[b,t,:] . Win_w[h,:] + Win_b[h]
// GEMM: [B*T, IN] x [IN, H].  Result written into the traj region of d_out.
// One wave = one 16-row M tile x 128 cols (8 n-tiles). 65536 waves total.
// ---------------------------------------------------------------------------
__global__ __launch_bounds__(256) void proj_kernel(
    const float* __restrict__ u, const float* __restrict__ Win_w,
    const float* __restrict__ Win_b, float* __restrict__ traj) {
    const int wave = (int)((blockIdx.x * blockDim.x + threadIdx.x) >> 5);
    const int lane = (int)(threadIdx.x & 31);
    const int hf   = lane >> 4;    // 0..1
    const int l    = lane & 15;    // 0..15

    const int mtile = wave >> 2;        // 0..16383
    const int ng    = wave & 3;         // 4 n-groups of 128 cols
    const int mr    = mtile * 16;
    const int nb0   = ng * 128;

    v8f acc[8];
    #pragma unroll
    for (int j = 0; j < 8; ++j) {
        const float bias = Win_b[nb0 + j * 16 + l];
        #pragma unroll
        for (int v = 0; v < 8; ++v) acc[j][v] = bias;
    }

    const size_t arow = (size_t)(mr + l) * INDIM;
    for (int kb = 0; kb < INDIM; kb += 4) {
        const int k0 = kb + 2 * hf;
        v2f a;
        a[0] = u[arow + k0];
        a[1] = u[arow + k0 + 1];
        #pragma unroll
        for (int j = 0; j < 8; ++j) {
            const int n = nb0 + j * 16 + l;     // B[k][n] = Win_w[n][k]
            v2f b;
            b[0] = Win_w[(size_t)n * INDIM + k0];
            b[1] = Win_w[(size_t)n * INDIM + k0 + 1];
            acc[j] = wmma_f32_16x16x4(a, b, acc[j]);
        }
    }

    #pragma unroll
    for (int j = 0; j < 8; ++j) {
        #pragma unroll
        for (int v = 0; v < 8; ++v) {
            const int row = v + hf * 8;
            traj[(size_t)(mr + row) * HDIM + nb0 + j * 16 + l] = acc[j][v];
        }
    }
}

// ---------------------------------------------------------------------------
// Kernel B: sequential recurrence. One block owns 16 batch rows for all T.
// x state lives in LDS (row stride 516 floats to avoid 64-bank conflicts).
// Per step: waves 0/1 do P = tanh(x)@M, Q = tanh(x)@N via WMMA (K=512),
// then all 8 waves do Hid = (P*Q*invH2) @ L^T (K=8) and the Euler update.
// traj holds the projection on entry and is overwritten with x_t in place.
// ---------------------------------------------------------------------------
#define XSTRIDE 516

__global__ __launch_bounds__(256) void recur_kernel(
    const float* __restrict__ x0, const float* __restrict__ noise,
    const float* __restrict__ Lm, const float* __restrict__ Mm,
    const float* __restrict__ Nm, float* __restrict__ traj,
    float* __restrict__ xfin) {
    __shared__ float xs[16 * XSTRIDE];   // state x[16][512] (padded)
    __shared__ float Pq[2][16 * 16];     // P and Q, R padded to 16
    __shared__ float Es[16 * 16];        // E = P*Q*invH2, cols 8..15 = 0

    const int tid  = (int)threadIdx.x;
    const int wave = tid >> 5;
    const int lane = tid & 31;
    const int hf   = lane >> 4;
    const int l    = lane & 15;
    const int b0   = (int)blockIdx.x * 16;

    // load x0 tile
    for (int i = tid; i < 16 * HDIM; i += 256) {
        const int r = i >> 9, c = i & 511;
        xs[r * XSTRIDE + c] = x0[(size_t)(b0 + r) * HDIM + c];
    }
    __syncthreads();

    for (int t = 0; t < TLEN; ++t) {
        // ---- P = tanh(x)@M (wave 0), Q = tanh(x)@N (wave 1) --------------
        if (wave < 2) {
            const float* S = (wave == 0) ? Mm : Nm;   // [H][R] row-major
            v8f acc = {};
            for (int kb = 0; kb < HDIM; kb += 4) {
                const int k0 = kb + 2 * hf;
                v2f a;
                a[0] = tanhf(xs[l * XSTRIDE + k0]);
                a[1] = tanhf(xs[l * XSTRIDE + k0 + 1]);
                v2f b;                                 // B[k][n] = S[k][n], n<8
                b[0] = (l < RDIM) ? S[(size_t)k0 * RDIM + l] : 0.0f;
                b[1] = (l < RDIM) ? S[(size_t)(k0 + 1) * RDIM + l] : 0.0f;
                acc = wmma_f32_16x16x4(a, b, acc);
            }
            #pragma unroll
            for (int v = 0; v < 8; ++v)
                Pq[wave][(v + hf * 8) * 16 + l] = acc[v];
        }
        __syncthreads();

        // ---- E = P * Q * invH2 (padded cols zeroed) ----------------------
        {
            const int row = tid >> 4, col = tid & 15;
            Es[row * 16 + col] =
                (col < RDIM)
                    ? Pq[0][row * 16 + col] * Pq[1][row * 16 + col] * INV_H2
                    : 0.0f;
        }
        __syncthreads();

        // ---- Hid = E @ L^T, fused Euler update. wave -> 64 H-columns -----
        #pragma unroll
        for (int jt = 0; jt < 4; ++jt) {
            const int nb = (wave * 4 + jt) * 16;
            v8f acc = {};
            #pragma unroll
            for (int kb = 0; kb < RDIM; kb += 4) {     // K = 8 (two WMMAs)
                const int k0 = kb + 2 * hf;            // 0..7, always < RDIM
                v2f a;
                a[0] = Es[l * 16 + k0];
                a[1] = Es[l * 16 + k0 + 1];
                v2f b;                                 // B[k][n] = L[n][k]
                b[0] = Lm[(size_t)(nb + l) * RDIM + k0];
                b[1] = Lm[(size_t)(nb + l) * RDIM + k0 + 1];
                acc = wmma_f32_16x16x4(a, b, acc);
            }
            #pragma unroll
            for (int v = 0; v < 8; ++v) {
                const int row = v + hf * 8;
                const int col = nb + l;
                const int bg  = b0 + row;
                const size_t ti = ((size_t)bg * TLEN + t) * HDIM + col;
                const float xo  = xs[row * XSTRIDE + col];
                const float inp = traj[ti];                                 // projection
                const float nz  = noise[((size_t)t * BSZ + bg) * HDIM + col];
                const float xn  = xo + NOISE_STD * nz + TAU * (-xo + acc[v] + inp);
                traj[ti] = xn;                                              // x_t
                xs[row * XSTRIDE + col] = xn;
            }
        }
        __syncthreads();
    }

    for (int i = tid; i < 16 * HDIM; i += 256) {
        const int r = i >> 9, c = i & 511;
        xfin[(size_t)(b0 + r) * HDIM + c] = xs[r * XSTRIDE + c];
    }
}

// ---------------------------------------------------------------------------
// Kernel C: readout  out[b,t,o] = tanh(traj[b,t,:]) . Wout_w[o,:] + Wout_b[o]
// GEMM: [B*T, H] x [H, OUT]. One wave = one 16-row M tile x all 64 cols.
// ---------------------------------------------------------------------------
__global__ __launch_bounds__(256) void readout_kernel(
    const float* __restrict__ traj, const float* __restrict__ Wout_w,
    const float* __restrict__ Wout_b, float* __restrict__ out) {
    const int wave = (int)((blockIdx.x * blockDim.x + threadIdx.x) >> 5);
    const int lane = (int)(threadIdx.x & 31);
    const int hf   = lane >> 4;
    const int l    = lane & 15;
    const int mr   = wave * 16;

    v8f acc[4];
    #pragma unroll
    for (int j = 0; j < 4; ++j) {
        const float bias = Wout_b[j * 16 + l];
        #pragma unroll
        for (int v = 0; v < 8; ++v) acc[j][v] = bias;
    }

    const size_t arow = (size_t)(mr + l) * HDIM;
    for (int kb = 0; kb < HDIM; kb += 4) {
        const int k0 = kb + 2 * hf;
        v2f a;
        a[0] = tanhf(traj[arow + k0]);
        a[1] = tanhf(traj[arow + k0 + 1]);
        #pragma unroll
        for (int j = 0; j < 4; ++j) {
            const int n = j * 16 + l;               // B[k][n] = Wout_w[n][k]
            v2f b;
            b[0] = Wout_w[(size_t)n * HDIM + k0];
            b[1] = Wout_w[(size_t)n * HDIM + k0 + 1];
            acc[j] = wmma_f32_16x16x4(a, b, acc[j]);
        }
    }

    #pragma unroll
    for (int j = 0; j < 4; ++j) {
        #pragma unroll
        for (int v = 0; v < 8; ++v) {
            const int row = v + hf * 8;
            out[(size_t)(mr + row) * OUTDIM + j * 16 + l] = acc[j][v];
        }
    }
}

// ---------------------------------------------------------------------------
extern "C" void kernel_launch(void* const* d_in, const int* in_sizes, int n_in,
                              void* d_out, int out_size, void* d_ws, size_t ws_size,
                              hipStream_t stream) {
    (void)in_sizes; (void)n_in; (void)out_size; (void)d_ws; (void)ws_size;

    const float* u      = (const float*)d_in[0];   // [B,T,IN]
    const float* x0     = (const float*)d_in[1];   // [B,H]
    const float* noise  = (const float*)d_in[2];   // [T,B,H]
    const float* Lm     = (const float*)d_in[3];   // [H,R]
    const float* Mm     = (const float*)d_in[4];   // [H,R]
    const float* Nm     = (const float*)d_in[5];   // [H,R]
    const float* Win_w  = (const float*)d_in[6];   // [H,IN]
    const float* Win_b  = (const float*)d_in[7];   // [H]
    const float* Wout_w = (const float*)d_in[8];   // [OUT,H]
    const float* Wout_b = (const float*)d_in[9];   // [OUT]

    float* out_output = (float*)d_out;                                   // [B,T,OUT]
    float* out_xfin   = out_output + (size_t)BSZ * TLEN * OUTDIM;        // [B,H]
    float* out_traj   = out_xfin + (size_t)BSZ * HDIM;                   // [B,T,H]

    // A: projection into traj region. 16384 M-tiles x 4 n-groups = 65536 waves.
    proj_kernel<<<8192, 256, 0, stream>>>(u, Win_w, Win_b, out_traj);

    // B: recurrence. 16 blocks x (16 batch rows each), 8 waves per block.
    recur_kernel<<<BSZ / 16, 256, 0, stream>>>(x0, noise, Lm, Mm, Nm,
                                               out_traj, out_xfin);

    // C: readout. 16384 waves, 8 per block.
    readout_kernel<<<2048, 256, 0, stream>>>(out_traj, Wout_w, Wout_b, out_output);
}